// HGCN_27075473834261
// MI455X (gfx1250) — compile-verified
//
#include <hip/hip_runtime.h>
#include <hip/hip_bf16.h>

typedef __attribute__((ext_vector_type(2))) float v2f;
typedef __attribute__((ext_vector_type(8))) float v8f;

#define N_SEQ_C 100000
#define N_LAB_C 64
#define E_CON_C 1600000
#define DDIM 128
#define BN_EPS_C 1e-5f

// ---------------- degree / scaling kernels ----------------

__global__ __launch_bounds__(256) void deg_kernel(const int* __restrict__ src,
                                                  const int* __restrict__ dst,
                                                  float* degout, float* degin, int E) {
  int e = blockIdx.x * 256 + threadIdx.x;
  if (e < E) {
    atomicAdd(&degout[src[e]], 1.0f);
    atomicAdd(&degin[dst[e]], 1.0f);
  }
}

__global__ __launch_bounds__(256) void cnt_kernel(const int* __restrict__ label,
                                                  float* cnt, int n) {
  __shared__ float s[64];
  if (threadIdx.x < 64) s[threadIdx.x] = 0.0f;
  __syncthreads();
  for (int i = blockIdx.x * 256 + threadIdx.x; i < n; i += gridDim.x * 256)
    atomicAdd(&s[label[i]], 1.0f);
  __syncthreads();
  if (threadIdx.x < 64) atomicAdd(&cnt[threadIdx.x], s[threadIdx.x]);
}

// deg -> deg^{-1/2}, zero-degree -> 0 (in place over a contiguous region)
__global__ __launch_bounds__(256) void isd_kernel(float* d, int n) {
  int i = blockIdx.x * 256 + threadIdx.x;
  if (i < n) {
    float v = d[i];
    d[i] = (v > 0.0f) ? rsqrtf(fmaxf(v, 1.0f)) : 0.0f;
  }
}

// ---------------- segment-sum of seq rows by label (64x128 LDS acc) ----------------

__global__ __launch_bounds__(256) void label_agg_kernel(const float* __restrict__ x,
                                                        const int* __restrict__ label,
                                                        float* lab_agg, int rows) {
  __shared__ float s[N_LAB_C * DDIM];  // 32 KB
  for (int i = threadIdx.x; i < N_LAB_C * DDIM; i += 256) s[i] = 0.0f;
  __syncthreads();
  int c = threadIdx.x & (DDIM - 1);
  int rp = threadIdx.x >> 7;  // 0/1
  for (int row = blockIdx.x * 2 + rp; row < rows; row += gridDim.x * 2)
    atomicAdd(&s[label[row] * DDIM + c], x[(size_t)row * DDIM + c]);
  __syncthreads();
  for (int i = threadIdx.x; i < N_LAB_C * DDIM; i += 256) atomicAdd(&lab_agg[i], s[i]);
}

// ---------------- edge scatter: agg[dst] += x[src] * isd_out[src] ----------------

__global__ __launch_bounds__(256) void scatter_con_kernel(const float* __restrict__ x,
                                                          const float* __restrict__ isd_out,
                                                          const int* __restrict__ src,
                                                          const int* __restrict__ dst,
                                                          float* agg, int E) {
  int idx = blockIdx.x * 256 + threadIdx.x;
  int e = idx >> 5;  // 32 lanes per edge, 4 floats each
  if (e >= E) return;
  int c4 = (idx & 31) * 4;
  int s = src[e], d = dst[e];
  float sc = isd_out[s];
  const float4 v = *(const float4*)(x + (size_t)s * DDIM + c4);
  float* ap = agg + (size_t)d * DDIM + c4;
  atomicAdd(ap + 0, v.x * sc);
  atomicAdd(ap + 1, v.y * sc);
  atomicAdd(ap + 2, v.z * sc);
  atomicAdd(ap + 3, v.w * sc);
}

// ---------------- small GEMM: out[64,N] = rowscale[l]*(A[64,128]@W) + b, opt relu ----------------

__global__ __launch_bounds__(256) void small_gemm_kernel(const float* __restrict__ A,
                                                         const float* __restrict__ rowscale,
                                                         const float* __restrict__ W,
                                                         const float* __restrict__ bias,
                                                         float* out, int N, int do_relu) {
  int idx = blockIdx.x * 256 + threadIdx.x;
  if (idx >= N_LAB_C * N) return;
  int l = idx / N, n = idx - l * N;
  float acc = 0.0f;
  for (int k = 0; k < DDIM; ++k) acc += A[l * DDIM + k] * W[k * N + n];
  acc = acc * rowscale[l] + bias[n];
  if (do_relu) acc = fmaxf(acc, 0.0f);
  out[idx] = acc;
}

// ---------------- big fused GEMM: f32 WMMA 16x16x4, K=128, N=128 ----------------
// out[m][n] = relu(0.5*( (isd_in[m]*agg[m]) @ W [n] + bias[n] + Y[label[m]][n] ))

__global__ __launch_bounds__(256) void wmma_gemm_fused(const float* __restrict__ A,
                                                       const float* __restrict__ isd_row,
                                                       const float* __restrict__ W,
                                                       const float* __restrict__ bias,
                                                       const float* __restrict__ Y,
                                                       const int* __restrict__ label,
                                                       float* __restrict__ out, int M) {
  __shared__ float sW[DDIM * DDIM];  // 64 KB weight stage in LDS
  for (int i = threadIdx.x; i < DDIM * DDIM; i += 256) sW[i] = W[i];
  __syncthreads();

  int wave = threadIdx.x >> 5;
  int lane = threadIdx.x & 31;
  int m0 = (blockIdx.x * 8 + wave) * 16;
  if (m0 >= M) return;  // wave-uniform: EXEC stays all-ones for WMMA

  int half = lane >> 4;       // 0: K+0/1 , 1: K+2/3
  int l16 = lane & 15;
  int arow = m0 + l16;        // A-fragment row for this lane (fixed over K)
  const float* Arow = A + (size_t)arow * DDIM;
  float sA = isd_row[arow];   // fold D_dst^{-1/2} into the A load

  v8f acc[8];
  const v8f vz = {0.f, 0.f, 0.f, 0.f, 0.f, 0.f, 0.f, 0.f};
#pragma unroll
  for (int t = 0; t < 8; ++t) acc[t] = vz;

  for (int k = 0; k < DDIM; k += 4) {
    int ka = k + 2 * half;
    v2f a;
    a.x = Arow[ka] * sA;
    a.y = Arow[ka + 1] * sA;
#pragma unroll
    for (int t = 0; t < 8; ++t) {
      int n = t * 16 + l16;
      v2f b;
      b.x = sW[ka * DDIM + n];
      b.y = sW[(ka + 1) * DDIM + n];
      acc[t] = __builtin_amdgcn_wmma_f32_16x16x4_f32(false, a, false, b,
                                                     (short)0, acc[t], false, false);
    }
  }

  // D layout: VGPR r -> row m0 + r + 8*half ; col = t*16 + (lane&15)
  int rowbase = m0 + 8 * half;
#pragma unroll
  for (int r = 0; r < 8; ++r) {
    int row = rowbase + r;
    const float* Yr = Y + (size_t)label[row] * DDIM;
#pragma unroll
    for (int t = 0; t < 8; ++t) {
      int n = t * 16 + l16;
      float v = 0.5f * (acc[t][r] + bias[n] + Yr[n]);
      out[(size_t)row * DDIM + n] = fmaxf(v, 0.0f);
    }
  }
}

// ---------------- BatchNorm (training mode, biased var) ----------------

__global__ __launch_bounds__(256) void bn_stats_kernel(const float* __restrict__ x,
                                                       float* sums, float* sumsq, int rows) {
  __shared__ float ls[256], lq[256];
  int c = threadIdx.x & (DDIM - 1);
  int rp = threadIdx.x >> 7;
  float s = 0.0f, q = 0.0f;
  for (int row = blockIdx.x * 2 + rp; row < rows; row += gridDim.x * 2) {
    float v = x[(size_t)row * DDIM + c];
    s += v;
    q += v * v;
  }
  ls[threadIdx.x] = s;
  lq[threadIdx.x] = q;
  __syncthreads();
  if (threadIdx.x < DDIM) {
    atomicAdd(&sums[c], ls[threadIdx.x] + ls[threadIdx.x + 128]);
    atomicAdd(&sumsq[c], lq[threadIdx.x] + lq[threadIdx.x + 128]);
  }
}

__global__ void bn_finalize_kernel(const float* sums, const float* sumsq,
                                   const float* g, const float* beta,
                                   float* scale, float* shift, float invN) {
  int c = threadIdx.x;
  if (c >= DDIM) return;
  float m = sums[c] * invN;
  float v = sumsq[c] * invN - m * m;
  float s = g[c] * rsqrtf(v + BN_EPS_C);
  scale[c] = s;
  shift[c] = beta[c] - m * s;
}

__global__ __launch_bounds__(256) void bn_apply_kernel(float* x, const float* scale,
                                                       const float* shift, size_t n) {
  for (size_t i = blockIdx.x * 256ull + threadIdx.x; i < n; i += (size_t)gridDim.x * 256ull) {
    int c = (int)(i & (DDIM - 1));
    x[i] = x[i] * scale[c] + shift[c];
  }
}

// BN over a small [64, C] matrix, one block
__global__ void bn_small_kernel(const float* __restrict__ x, const float* g,
                                const float* beta, float* out, int rows, int C) {
  int c = threadIdx.x;
  if (c >= C) return;
  float s = 0.0f;
  for (int r = 0; r < rows; ++r) s += x[r * C + c];
  float m = s / (float)rows;
  float q = 0.0f;
  for (int r = 0; r < rows; ++r) {
    float d = x[r * C + c] - m;
    q += d * d;
  }
  float inv = rsqrtf(q / (float)rows + BN_EPS_C);
  for (int r = 0; r < rows; ++r)
    out[r * C + c] = (x[r * C + c] - m) * inv * g[c] + beta[c];
}

// ---------------- host orchestration ----------------

extern "C" void kernel_launch(void* const* d_in, const int* in_sizes, int n_in,
                              void* d_out, int out_size, void* d_ws, size_t ws_size,
                              hipStream_t stream) {
  const float* x_seq   = (const float*)d_in[0];
  const float* x_label = (const float*)d_in[1];
  const float* W1_bel = (const float*)d_in[2];  const float* b1_bel = (const float*)d_in[3];
  const float* W1_inc = (const float*)d_in[4];  const float* b1_inc = (const float*)d_in[5];
  const float* W1_con = (const float*)d_in[6];  const float* b1_con = (const float*)d_in[7];
  const float* g1s = (const float*)d_in[8];     const float* beta1s = (const float*)d_in[9];
  const float* g1l = (const float*)d_in[10];    const float* beta1l = (const float*)d_in[11];
  const float* W2_bel = (const float*)d_in[12]; const float* b2_bel = (const float*)d_in[13];
  const float* W2_inc = (const float*)d_in[14]; const float* b2_inc = (const float*)d_in[15];
  const float* W2_con = (const float*)d_in[16]; const float* b2_con = (const float*)d_in[17];
  const float* g2s = (const float*)d_in[18];    const float* beta2s = (const float*)d_in[19];
  const float* g2l = (const float*)d_in[20];    const float* beta2l = (const float*)d_in[21];
  const int* bel_label = (const int*)d_in[22];
  const int* con_src   = (const int*)d_in[23];
  const int* con_dst   = (const int*)d_in[24];

  float* ws = (float*)d_ws;
  // workspace layout (float offsets)
  float* degout  = ws + 0;        // 100000  -> isd_out
  float* degin   = ws + 100000;   // 100000  -> isd_in
  float* cnt_isd = ws + 200000;   // 64
  float* lab_agg = ws + 200064;   // 8192
  float* h_l     = ws + 208256;   // 8192  (label features between layers)
  float* Ybuf    = ws + 216448;   // 8192
  float* sums    = ws + 224640;   // 128
  float* sumsq   = ws + 224768;   // 128
  float* scale   = ws + 224896;   // 128
  float* shift   = ws + 225024;   // 128
  float* h2_l    = ws + 225152;   // 4096
  float* agg     = ws + 229376;   // 12,800,000

  float* out_seq = (float*)d_out;                 // [100000,128]; also layer-1 hidden buffer
  float* out_lab = (float*)d_out + (size_t)N_SEQ_C * DDIM;  // [64,64]

  // ---- degrees (shared by both layers) ----
  hipMemsetAsync(ws, 0, (size_t)200064 * sizeof(float), stream);
  deg_kernel<<<(E_CON_C + 255) / 256, 256, 0, stream>>>(con_src, con_dst, degout, degin, E_CON_C);
  cnt_kernel<<<128, 256, 0, stream>>>(bel_label, cnt_isd, N_SEQ_C);
  isd_kernel<<<(200064 + 255) / 256, 256, 0, stream>>>(ws, 200064);

  auto run_layer = [&](const float* in_seq, const float* in_lab,
                       const float* Wb, const float* bb, int Nbel, float* lab_out,
                       const float* Wi, const float* bi,
                       const float* Wc, const float* bc, float* seq_out) {
    hipMemsetAsync(agg, 0, (size_t)N_SEQ_C * DDIM * sizeof(float), stream);
    hipMemsetAsync(lab_agg, 0, (size_t)N_LAB_C * DDIM * sizeof(float), stream);
    hipMemsetAsync(sums, 0, 256 * sizeof(float), stream);
    // belongs_to: segment-sum + tiny GEMM (post-relu, pre-BN label features)
    label_agg_kernel<<<400, 256, 0, stream>>>(in_seq, bel_label, lab_agg, N_SEQ_C);
    small_gemm_kernel<<<(N_LAB_C * Nbel + 255) / 256, 256, 0, stream>>>(
        lab_agg, cnt_isd, Wb, bb, lab_out, Nbel, 1);
    // including: 64-row lookup table Y
    small_gemm_kernel<<<(N_LAB_C * DDIM + 255) / 256, 256, 0, stream>>>(
        in_lab, cnt_isd, Wi, bi, Ybuf, DDIM, 0);
    // connected_to: scatter then fused WMMA GEMM (+Y[label], *0.5, relu)
    scatter_con_kernel<<<(E_CON_C * 32 + 255) / 256, 256, 0, stream>>>(
        in_seq, degout, con_src, con_dst, agg, E_CON_C);
    int tiles = N_SEQ_C / 16;  // 6250
    wmma_gemm_fused<<<(tiles + 7) / 8, 256, 0, stream>>>(
        agg, degin, Wc, bc, Ybuf, bel_label, seq_out, N_SEQ_C);
  };

  // ================= layer 1 =================
  run_layer(x_seq, x_label, W1_bel, b1_bel, DDIM, h_l, W1_inc, b1_inc, W1_con, b1_con, out_seq);
  // BN seq (in place on out_seq == layer-1 hidden)
  bn_stats_kernel<<<512, 256, 0, stream>>>(out_seq, sums, sumsq, N_SEQ_C);
  bn_finalize_kernel<<<1, 128, 0, stream>>>(sums, sumsq, g1s, beta1s, scale, shift,
                                            1.0f / (float)N_SEQ_C);
  bn_apply_kernel<<<2048, 256, 0, stream>>>(out_seq, scale, shift, (size_t)N_SEQ_C * DDIM);
  // BN label (in place)
  bn_small_kernel<<<1, 128, 0, stream>>>(h_l, g1l, beta1l, h_l, N_LAB_C, DDIM);

  // ================= layer 2 =================
  // reads of out_seq (scatter, label_agg) are stream-ordered before the GEMM overwrites it
  run_layer(out_seq, h_l, W2_bel, b2_bel, 64, h2_l, W2_inc, b2_inc, W2_con, b2_con, out_seq);
  bn_stats_kernel<<<512, 256, 0, stream>>>(out_seq, sums, sumsq, N_SEQ_C);
  bn_finalize_kernel<<<1, 128, 0, stream>>>(sums, sumsq, g2s, beta2s, scale, shift,
                                            1.0f / (float)N_SEQ_C);
  bn_apply_kernel<<<2048, 256, 0, stream>>>(out_seq, scale, shift, (size_t)N_SEQ_C * DDIM);
  bn_small_kernel<<<1, 64, 0, stream>>>(h2_l, g2l, beta2l, out_lab, N_LAB_C, 64);
}